// QuantumEmbedding_75728863363349
// MI455X (gfx1250) — compile-verified
//
#include <hip/hip_runtime.h>
#include <hip/hip_bf16.h>

#define NQ      6
#define NLAYER  3
#define QDIM    64          // 1 << NQ
#define VOCAB   32000
#define EMBED   768
#define NTOK    (16 * 4096) // B * S

typedef float v2f __attribute__((ext_vector_type(2)));
typedef float v8f __attribute__((ext_vector_type(8)));

// ---------------------------------------------------------------------------
// Kernel 1: per-vocab-entry 6-qubit statevector simulation -> ev[VOCAB][6].
// One thread = one vocab id; 64 complex amplitudes fully in registers with
// compile-time indices (everything unrolled).
// ---------------------------------------------------------------------------
__global__ __launch_bounds__(128)
void qev_kernel(const float* __restrict__ qp, float* __restrict__ evb) {
    int v = blockIdx.x * blockDim.x + threadIdx.x;
    if (v >= VOCAB) return;

    float sr[QDIM], si[QDIM];
#pragma unroll
    for (int i = 0; i < QDIM; ++i) { sr[i] = 0.f; si[i] = 0.f; }
    sr[0] = 1.f;

    const float* p = qp + (size_t)v * (NLAYER * NQ * 2);

#pragma unroll
    for (int layer = 0; layer < NLAYER; ++layer) {
#pragma unroll
        for (int q = 0; q < NQ; ++q) {
            float ty = 0.5f * p[(layer * NQ + q) * 2 + 0];
            float tz = 0.5f * p[(layer * NQ + q) * 2 + 1];
            float cy, sy, cz, sz;
            __sincosf(ty, &sy, &cy);
            __sincosf(tz, &sz, &cz);
            const int m = 1 << (NQ - 1 - q);   // qubit-q bit mask
#pragma unroll
            for (int i = 0; i < QDIM; ++i) {
                if (i & m) continue;
                const int j = i | m;
                float a0r = sr[i], a0i = si[i];
                float a1r = sr[j], a1i = si[j];
                // RY: a0' = c a0 - s a1 ; a1' = s a0 + c a1
                float b0r = cy * a0r - sy * a1r, b0i = cy * a0i - sy * a1i;
                float b1r = sy * a0r + cy * a1r, b1i = sy * a0i + cy * a1i;
                // RZ: a0 *= e^{-i t/2} ; a1 *= e^{+i t/2}
                sr[i] = b0r * cz + b0i * sz;  si[i] = b0i * cz - b0r * sz;
                sr[j] = b1r * cz - b1i * sz;  si[j] = b1i * cz + b1r * sz;
            }
        }
        if (layer < NLAYER - 1) {
            // CNOT ring: (0,1),(1,2),(2,3),(3,4),(4,5),(5,0)
#pragma unroll
            for (int e = 0; e < NQ; ++e) {
                const int cm = 1 << (NQ - 1 - e);
                const int tm = 1 << (NQ - 1 - ((e + 1) % NQ));
#pragma unroll
                for (int i = 0; i < QDIM; ++i) {
                    if ((i & cm) && !(i & tm)) {
                        const int j = i ^ tm;
                        float tr = sr[i]; sr[i] = sr[j]; sr[j] = tr;
                        float ti = si[i]; si[i] = si[j]; si[j] = ti;
                    }
                }
            }
        }
    }

    float ev[NQ];
#pragma unroll
    for (int q = 0; q < NQ; ++q) ev[q] = 0.f;
#pragma unroll
    for (int i = 0; i < QDIM; ++i) {
        float pr = sr[i] * sr[i] + si[i] * si[i];
#pragma unroll
        for (int q = 0; q < NQ; ++q)
            ev[q] += ((i >> (NQ - 1 - q)) & 1) ? -pr : pr;
    }
#pragma unroll
    for (int q = 0; q < NQ; ++q) evb[(size_t)v * NQ + q] = ev[q];
}

// ---------------------------------------------------------------------------
// Kernel 2: gather ev by token id + 16x16-tiled projection via
// V_WMMA_F32_16X16X4_F32 (K=6 padded to 8 -> two WMMA steps).
//  - post_w / post_b staged in LDS in WMMA-ready layout (ds_load_b64 frags)
//  - bias used as the C-matrix initializer (no post-add)
//  - no EXEC-divergent loads: address-select for B, mul-mask for A padding
// One wave owns 16 tokens, sweeps 48 column tiles; 8 waves/block, 512 blocks.
// ---------------------------------------------------------------------------
__global__ __launch_bounds__(256)
void emb_kernel(const int*   __restrict__ tok,
                const float* __restrict__ evb,
                const float* __restrict__ pw,   // [EMBED][6] row-major
                const float* __restrict__ pb,   // [EMBED]
                float*       __restrict__ out)  // [NTOK][EMBED]
{
    // sW[col*8 + k] = post_w[col][k] (k<6), padding k=6,7 don't-care (zeroed)
    __shared__ float sW[EMBED * 8];
    __shared__ float sB[EMBED];

    const int tid = threadIdx.x;
    for (int idx = tid; idx < EMBED * 8; idx += 256) {
        const int col = idx >> 3;
        const int k   = idx & 7;
        sW[idx] = (k < NQ) ? pw[col * NQ + k] : 0.f;
    }
    for (int idx = tid; idx < EMBED; idx += 256)
        sB[idx] = pb[idx];
    __syncthreads();

    const int lane    = tid & 31;
    const int wave    = tid >> 5;
    const int tile    = blockIdx.x * 8 + wave;   // 16-token tile (4096 total)
    const int rowbase = tile * 16;
    const int mr      = lane & 15;               // A-row / B,C-column in tile
    const bool hi     = lane >= 16;

    // --- A matrix: 16 tokens x K(0..7) f32, ISA 16x4 layout ----------------
    // WMMA#1: lanes 0-15 v={K0,K1}; lanes 16-31 v={K2,K3}
    // WMMA#2: lanes 0-15 v={K4,K5}; lanes 16-31 v={0,0} (K=6,7 zero-pad)
    const int id = tok[rowbase + mr];
    const float* e = evb + (size_t)id * NQ;
    const int k0 = hi ? 2 : 0;
    const float e0 = e[k0], e1 = e[k0 + 1];      // unconditional loads
    const float e4 = e[4],  e5 = e[5];
    const float zm = hi ? 0.f : 1.f;             // mask, not a branch
    v2f a0, a1;
    a0[0] = e0;      a0[1] = e1;
    a1[0] = e4 * zm; a1[1] = e5 * zm;

    const int b1off = hi ? 0 : 4;  // hi-lane B rows K=6,7 are don't-care (A=0)

    for (int n = 0; n < EMBED / 16; ++n) {
        const int col = n * 16 + mr;

        const v2f b0 = *(const v2f*)&sW[col * 8 + k0];     // ds_load_b64
        const v2f b1 = *(const v2f*)&sW[col * 8 + b1off];  // ds_load_b64
        const float bias = sB[col];

        // C(m,n) = bias[n]; every one of this lane's 8 C regs shares col n
        v8f c;
#pragma unroll
        for (int r = 0; r < 8; ++r) c[r] = bias;

        c = __builtin_amdgcn_wmma_f32_16x16x4_f32(false, a0, false, b0,
                                                  (short)0, c, false, false);
        c = __builtin_amdgcn_wmma_f32_16x16x4_f32(false, a1, false, b1,
                                                  (short)0, c, false, false);

        // C layout: VGPR r -> row M=r (lanes 0-15) / M=r+8 (lanes 16-31)
        const int mo = hi ? 8 : 0;
        float* o = out + (size_t)(rowbase + mo) * EMBED + col;
#pragma unroll
        for (int r = 0; r < 8; ++r)
            o[(size_t)r * EMBED] = c[r];
    }
}

extern "C" void kernel_launch(void* const* d_in, const int* in_sizes, int n_in,
                              void* d_out, int out_size, void* d_ws, size_t ws_size,
                              hipStream_t stream) {
    const int*   token_ids = (const int*)  d_in[0];   // [16][4096] int32
    const float* qparams   = (const float*)d_in[1];   // [32000][3][6][2]
    const float* post_w    = (const float*)d_in[2];   // [768][6]
    const float* post_b    = (const float*)d_in[3];   // [768]
    float*       out       = (float*)d_out;           // [16][4096][768]
    float*       evbuf     = (float*)d_ws;            // [32000][6] = 768 KB

    // 1) per-vocab expectation values (one-shot, L2-resident result)
    qev_kernel<<<(VOCAB + 127) / 128, 128, 0, stream>>>(qparams, evbuf);

    // 2) gather + WMMA projection (4096 row tiles / 8 waves per block)
    emb_kernel<<<512, 256, 0, stream>>>(token_ids, evbuf, post_w, post_b, out);
}